// CustomLoss_57767310131732
// MI455X (gfx1250) — compile-verified
//
#include <hip/hip_runtime.h>
#include <cmath>

typedef __attribute__((ext_vector_type(2))) float v2f;
typedef __attribute__((ext_vector_type(8))) float v8f;

// ---------------- block reduction helper (wave32) ----------------
__device__ __forceinline__ float block_reduce_sum(float v) {
    __shared__ float red[32];
    int lane = threadIdx.x & 31;
    int wave = threadIdx.x >> 5;
#pragma unroll
    for (int o = 16; o; o >>= 1) v += __shfl_down(v, o, 32);
    if (lane == 0) red[wave] = v;
    __syncthreads();
    int nwaves = blockDim.x >> 5;
    v = (threadIdx.x < nwaves) ? red[threadIdx.x] : 0.0f;
    if (wave == 0) {
#pragma unroll
        for (int o = 16; o; o >>= 1) v += __shfl_down(v, o, 32);
    }
    return v; // valid in thread 0
}

// ---------------- init accumulators ----------------
__global__ void init_kernel(float* ws) {
    if (threadIdx.x < 3) ws[threadIdx.x] = 0.0f;
}

// ---------------- sum((a-b)^2), float4 grid-stride ----------------
__global__ void ssq_diff_kernel(const float* __restrict__ a,
                                const float* __restrict__ b,
                                float* __restrict__ accum, size_t n) {
    size_t tid = (size_t)blockIdx.x * blockDim.x + threadIdx.x;
    size_t stride = (size_t)gridDim.x * blockDim.x;
    size_t n4 = n >> 2;
    const float4* a4 = (const float4*)a;
    const float4* b4 = (const float4*)b;
    float s = 0.0f;
    for (size_t i = tid; i < n4; i += stride) {
        float4 x = a4[i], y = b4[i];
        float d0 = x.x - y.x, d1 = x.y - y.y, d2 = x.z - y.z, d3 = x.w - y.w;
        s = fmaf(d0, d0, s); s = fmaf(d1, d1, s);
        s = fmaf(d2, d2, s); s = fmaf(d3, d3, s);
    }
    for (size_t i = (n4 << 2) + tid; i < n; i += stride) {
        float d = a[i] - b[i]; s = fmaf(d, d, s);
    }
    s = block_reduce_sum(s);
    if (threadIdx.x == 0) atomicAdd(accum, s);
}

// ---------------- sum(relu(p)^2), float4 grid-stride ----------------
__global__ void ssq_relu_kernel(const float* __restrict__ p,
                                float* __restrict__ accum, size_t n) {
    size_t tid = (size_t)blockIdx.x * blockDim.x + threadIdx.x;
    size_t stride = (size_t)gridDim.x * blockDim.x;
    size_t n4 = n >> 2;
    const float4* p4 = (const float4*)p;
    float s = 0.0f;
    for (size_t i = tid; i < n4; i += stride) {
        float4 x = p4[i];
        float r0 = fmaxf(x.x, 0.0f), r1 = fmaxf(x.y, 0.0f);
        float r2 = fmaxf(x.z, 0.0f), r3 = fmaxf(x.w, 0.0f);
        s = fmaf(r0, r0, s); s = fmaf(r1, r1, s);
        s = fmaf(r2, r2, s); s = fmaf(r3, r3, s);
    }
    for (size_t i = (n4 << 2) + tid; i < n; i += stride) {
        float r = fmaxf(p[i], 0.0f); s = fmaf(r, r, s);
    }
    s = block_reduce_sum(s);
    if (threadIdx.x == 0) atomicAdd(accum, s);
}

// ---------------- ||S[idx[row]]||^2 for gathered rows ----------------
__global__ void row_norms_kernel(const float* __restrict__ S,
                                 const int* __restrict__ iidx,
                                 const int* __restrict__ jidx,
                                 float* __restrict__ ni, float* __restrict__ nj,
                                 int D, int Npair) {
    int which = (blockIdx.x >= (unsigned)Npair) ? 1 : 0;
    int row = blockIdx.x - which * Npair;
    int g = which ? jidx[row] : iidx[row];
    const float* p = S + (size_t)g * D;
    float s = 0.0f;
    for (int k = threadIdx.x; k < D; k += blockDim.x) {
        float x = p[k]; s = fmaf(x, x, s);
    }
    s = block_reduce_sum(s);
    if (threadIdx.x == 0) (which ? nj : ni)[row] = s;
}

// ---------------- WMMA Gram kernel: G = Si * Sj^T, then penalty ----------------
// One wave per 16x16 output tile. A = Si tile (16xK), B = Sj^T tile (Kx16).
// f32 16x16x4 layouts (ISA 7.12.2):
//   A: lane holds row M=lane%16; lanes 0-15 K={k0,k0+1}, lanes 16-31 K={k0+2,k0+3}
//   B: lane holds col N=lane%16; same K split
//   C: VGPR v holds M = v + 8*(lane/16), N = lane%16
__global__ void gram_penalty_kernel(const float* __restrict__ S,
                                    const float* __restrict__ P,
                                    const int* __restrict__ iidx,
                                    const int* __restrict__ jidx,
                                    const float* __restrict__ ni,
                                    const float* __restrict__ nj,
                                    float* __restrict__ accum,
                                    int D, int K, int tcols) {
    int wave = threadIdx.x >> 5;
    int lane = threadIdx.x & 31;
    int tile = blockIdx.x * (blockDim.x >> 5) + wave;
    int ti = tile / tcols;
    int tj = tile - ti * tcols;
    int r  = lane & 15;   // A-row / B-col for this lane
    int kh = lane >> 4;   // which K half-pair this lane supplies

    int gi_row = iidx[ti * 16 + r];
    int gj_row = jidx[tj * 16 + r];
    const float* arow = S + (size_t)gi_row * D;
    const float* brow = S + (size_t)gj_row * D;

    v8f c0 = {};
    v8f c1 = {};
    int koff = 2 * kh;
#pragma unroll 4
    for (int k0 = 0; k0 < D; k0 += 8) {
        v2f a0 = *(const v2f*)(arow + k0 + koff);
        v2f b0 = *(const v2f*)(brow + k0 + koff);
        v2f a1 = *(const v2f*)(arow + k0 + 4 + koff);
        v2f b1 = *(const v2f*)(brow + k0 + 4 + koff);
        c0 = __builtin_amdgcn_wmma_f32_16x16x4_f32(false, a0, false, b0,
                                                   (short)0, c0, false, false);
        c1 = __builtin_amdgcn_wmma_f32_16x16x4_f32(false, a1, false, b1,
                                                   (short)0, c1, false, false);
    }

    float psum = 0.0f;
#pragma unroll
    for (int v = 0; v < 8; v++) {
        float g = c0[v] + c1[v];
        int m = v + 8 * kh;            // local row in tile
        int i_loc = ti * 16 + m;       // 0..Npair-1
        int j_loc = tj * 16 + r;
        float d2 = ni[i_loc] + nj[j_loc] - 2.0f * g;
        float nrm = (d2 > 0.0f) ? sqrtf(d2) : 0.0f;
        int gi = iidx[i_loc];
        int gj = jidx[j_loc];
        float pij = P[(size_t)gi * K + gj];
        pij = fmaxf(pij, 0.0f);
        psum = fmaf(pij, nrm, psum);
    }
#pragma unroll
    for (int o = 16; o; o >>= 1) psum += __shfl_down(psum, o, 32);
    if (lane == 0) atomicAdd(accum, psum);
}

// ---------------- finalize ----------------
__global__ void finalize_kernel(const float* __restrict__ ws,
                                const float* __restrict__ lamb,
                                float* __restrict__ out) {
    if (threadIdx.x == 0 && blockIdx.x == 0) {
        out[0] = sqrtf(ws[0]) + lamb[0] * (sqrtf(ws[1]) + ws[2]);
    }
}

extern "C" void kernel_launch(void* const* d_in, const int* in_sizes, int n_in,
                              void* d_out, int out_size, void* d_ws, size_t ws_size,
                              hipStream_t stream) {
    const float* actual = (const float*)d_in[0];
    const float* pred   = (const float*)d_in[1];
    const float* lamb   = (const float*)d_in[2];
    const float* P      = (const float*)d_in[3];
    const float* S      = (const float*)d_in[4];
    const int*   iidx   = (const int*)d_in[5];
    const int*   jidx   = (const int*)d_in[6];
    float* out = (float*)d_out;

    size_t nData = (size_t)in_sizes[0];          // 4096*4096
    size_t nP    = (size_t)in_sizes[3];          // K*K
    int K = (int)(std::sqrt((double)nP) + 0.5);  // 2048
    int D = in_sizes[4] / K;                     // 1024
    int Npair = in_sizes[5];                     // 128

    float* ws = (float*)d_ws;   // [0]=ss_data [1]=ss_P [2]=param_penalty
    float* ni = ws + 16;
    float* nj = ni + Npair;

    init_kernel<<<1, 32, 0, stream>>>(ws);
    ssq_diff_kernel<<<2048, 256, 0, stream>>>(actual, pred, ws + 0, nData);
    ssq_relu_kernel<<<1024, 256, 0, stream>>>(P, ws + 1, nP);
    row_norms_kernel<<<2 * Npair, 256, 0, stream>>>(S, iidx, jidx, ni, nj, D, Npair);

    int tcols = Npair / 16;                 // 8
    int tiles = tcols * tcols;              // 64 waves
    int waves_per_block = 8;                // 256 threads
    gram_penalty_kernel<<<tiles / waves_per_block, waves_per_block * 32, 0, stream>>>(
        S, P, iidx, jidx, ni, nj, ws + 2, D, K, tcols);

    finalize_kernel<<<1, 32, 0, stream>>>(ws, lamb, out);
}